// BipartiteLayer_29892972380766
// MI455X (gfx1250) — compile-verified
//
#include <hip/hip_runtime.h>
#include <hip/hip_bf16.h>

#define N_INTT   100000
#define N_MVTX   100000
#define N_EDGES  300000
#define IN_DIM   64
#define FEAT_DIM 128          // 2*OUT_DIM
#define EDGE_DIM 256          // 2*FEAT_DIM
#define H_DIM    704          // 64 + 128 + 512
#define OUT_DIM  64

typedef __attribute__((ext_vector_type(2))) float v2f;
typedef __attribute__((ext_vector_type(8))) float v8f;

// ---------------------------------------------------------------------------
// Zero scratch (pool sums / maxes / counts). Graph-capture safe.
// ---------------------------------------------------------------------------
__global__ void zero_kernel(float* __restrict__ p, size_t n) {
    size_t i = (size_t)blockIdx.x * blockDim.x + threadIdx.x;
    size_t stride = (size_t)gridDim.x * blockDim.x;
    for (; i < n; i += stride) p[i] = 0.0f;
}

// ---------------------------------------------------------------------------
// Input transform: XP[N,128] = X[N,64] @ W[64,128] + b     (fp32 WMMA 16x16x4)
// Block = 256 threads = 8 waves; each wave owns one 16-col tile of a 16-row
// block. A-frag: lane r=lane%16 holds row r; VGPRs {0,1} hold K{0,1} for
// lanes 0-15 and K{2,3} for lanes 16-31 (per ISA 32-bit A layout).
// ---------------------------------------------------------------------------
__global__ __launch_bounds__(256)
void in_gemm(const float* __restrict__ X, const float* __restrict__ W,
             const float* __restrict__ bias, float* __restrict__ XP) {
    const int lane    = threadIdx.x & 31;
    const int wv      = threadIdx.x >> 5;       // 0..7 -> column tile
    const int r       = lane & 15;
    const int hi      = lane >> 4;              // lane half
    const int rowbase = blockIdx.x * 16;
    const int coltile = wv * 16;

    const float* __restrict__ Arow = X + (size_t)(rowbase + r) * IN_DIM;

    v8f c = {};
#pragma unroll
    for (int k0 = 0; k0 < IN_DIM; k0 += 4) {
        const int k = k0 + 2 * hi;
        v2f a, b;
        a[0] = Arow[k];
        a[1] = Arow[k + 1];
        b[0] = W[(size_t)k       * FEAT_DIM + coltile + r];
        b[1] = W[(size_t)(k + 1) * FEAT_DIM + coltile + r];
        c = __builtin_amdgcn_wmma_f32_16x16x4_f32(false, a, false, b,
                                                  (short)0, c, false, false);
    }

    const int col = coltile + r;
    const float bb = bias[col];
#pragma unroll
    for (int i = 0; i < 8; ++i) {
        const int row = rowbase + i + 8 * hi;   // C/D: VGPR i -> M=i / M=8+i
        XP[(size_t)row * FEAT_DIM + col] = c[i] + bb;
    }
}

// ---------------------------------------------------------------------------
// Edge kernel: one edge per wave32. Each lane covers 4 channels of the intt
// half and 4 of the mvtx half (b128 loads). Score reduced across the wave,
// att = exp(-|score+b|); scatter add (mean numerator) + uint-bitpattern
// atomic max (exact for positives; max buffer is 0-initialized and the
// reference clamps the segment max at 0, so negatives can be skipped).
// ---------------------------------------------------------------------------
__global__ __launch_bounds__(256)
void edge_kernel(const float* __restrict__ xp_i, const float* __restrict__ xp_m,
                 const int* __restrict__ eidx,
                 const float* __restrict__ Ws, const float* __restrict__ bs,
                 float* __restrict__ sum_i, float* __restrict__ max_i,
                 float* __restrict__ cnt_i,
                 float* __restrict__ sum_m, float* __restrict__ max_m,
                 float* __restrict__ cnt_m) {
    const int lane = threadIdx.x & 31;
    const int wv   = threadIdx.x >> 5;
    const int e    = blockIdx.x * 8 + wv;       // grid sized exactly
    const int s    = eidx[e];                   // intt endpoint
    const int t    = eidx[N_EDGES + e];         // mvtx endpoint

    const int ch0 = lane * 4;
    const float4 xi = *(const float4*)(xp_i + (size_t)s * FEAT_DIM + ch0);
    const float4 xm = *(const float4*)(xp_m + (size_t)t * FEAT_DIM + ch0);
    const float4 w0 = *(const float4*)(Ws + ch0);
    const float4 w1 = *(const float4*)(Ws + FEAT_DIM + ch0);

    float p = xi.x * w0.x + xi.y * w0.y + xi.z * w0.z + xi.w * w0.w
            + xm.x * w1.x + xm.y * w1.y + xm.z * w1.z + xm.w * w1.w;
#pragma unroll
    for (int off = 16; off >= 1; off >>= 1) p += __shfl_xor(p, off, 32);

    const float att = __expf(-fabsf(p + bs[0]));

    float v[8] = { xi.x * att, xi.y * att, xi.z * att, xi.w * att,
                   xm.x * att, xm.y * att, xm.z * att, xm.w * att };

    float* __restrict__ si = sum_i + (size_t)s * EDGE_DIM;
    float* __restrict__ sm = sum_m + (size_t)t * EDGE_DIM;
    float* __restrict__ mi = max_i + (size_t)s * EDGE_DIM;
    float* __restrict__ mm = max_m + (size_t)t * EDGE_DIM;

#pragma unroll
    for (int cix = 0; cix < 4; ++cix) {
        const int ca = ch0 + cix;               // intt-half channel
        const int cb = FEAT_DIM + ch0 + cix;    // mvtx-half channel
        atomicAdd(si + ca, v[cix]);
        atomicAdd(sm + ca, v[cix]);
        atomicAdd(si + cb, v[4 + cix]);
        atomicAdd(sm + cb, v[4 + cix]);
        if (v[cix] > 0.0f) {
            const unsigned u = __float_as_uint(v[cix]);
            atomicMax((unsigned*)(mi + ca), u);
            atomicMax((unsigned*)(mm + ca), u);
        }
        if (v[4 + cix] > 0.0f) {
            const unsigned u = __float_as_uint(v[4 + cix]);
            atomicMax((unsigned*)(mi + cb), u);
            atomicMax((unsigned*)(mm + cb), u);
        }
    }
    if (lane == 0) {
        atomicAdd(cnt_i + s, 1.0f);
        atomicAdd(cnt_m + t, 1.0f);
    }
}

// ---------------------------------------------------------------------------
// Output GEMM with fused concat: H = [x(64) | xp(128) | sum/cnt(256) | max(256)]
// OUT[N,64] = relu(H @ W[704,64] + b).  Block = 128 threads = 4 waves, one
// 16-col tile each; 176 fp32 WMMA k-steps. Segment boundaries (64/192/448)
// are 4-aligned so source selection is uniform per k-step.
// ---------------------------------------------------------------------------
__global__ __launch_bounds__(128)
void out_gemm(const float* __restrict__ X,   const float* __restrict__ XP,
              const float* __restrict__ SUM, const float* __restrict__ MAXB,
              const float* __restrict__ CNT,
              const float* __restrict__ W,   const float* __restrict__ bias,
              float* __restrict__ OUT) {
    const int lane    = threadIdx.x & 31;
    const int wv      = threadIdx.x >> 5;       // 0..3
    const int r       = lane & 15;
    const int hi      = lane >> 4;
    const int rowbase = blockIdx.x * 16;
    const int row     = rowbase + r;
    const int coltile = wv * 16;

    const float invc = 1.0f / fmaxf(CNT[row], 1.0f);
    const float* __restrict__ xrow  = X    + (size_t)row * IN_DIM;
    const float* __restrict__ xprow = XP   + (size_t)row * FEAT_DIM;
    const float* __restrict__ srow  = SUM  + (size_t)row * EDGE_DIM;
    const float* __restrict__ mrow  = MAXB + (size_t)row * EDGE_DIM;

    v8f c = {};
#pragma unroll 4
    for (int k0 = 0; k0 < H_DIM; k0 += 4) {
        const int k = k0 + 2 * hi;
        v2f a, b;
        if (k0 < 64) {
            a[0] = xrow[k];                a[1] = xrow[k + 1];
        } else if (k0 < 192) {
            a[0] = xprow[k - 64];          a[1] = xprow[k - 63];
        } else if (k0 < 448) {
            a[0] = srow[k - 192] * invc;   a[1] = srow[k - 191] * invc;
        } else {
            a[0] = mrow[k - 448];          a[1] = mrow[k - 447];
        }
        b[0] = W[(size_t)k       * OUT_DIM + coltile + r];
        b[1] = W[(size_t)(k + 1) * OUT_DIM + coltile + r];
        c = __builtin_amdgcn_wmma_f32_16x16x4_f32(false, a, false, b,
                                                  (short)0, c, false, false);
    }

    const int col = coltile + r;
    const float bb = bias[col];
#pragma unroll
    for (int i = 0; i < 8; ++i) {
        const int orow = rowbase + i + 8 * hi;
        OUT[(size_t)orow * OUT_DIM + col] = fmaxf(c[i] + bb, 0.0f);
    }
}

// ---------------------------------------------------------------------------
extern "C" void kernel_launch(void* const* d_in, const int* in_sizes, int n_in,
                              void* d_out, int out_size, void* d_ws, size_t ws_size,
                              hipStream_t stream) {
    (void)in_sizes; (void)n_in; (void)out_size; (void)ws_size;

    // setup_inputs() order
    const float* x_intt   = (const float*)d_in[2];
    const float* x_mvtx   = (const float*)d_in[3];
    const int*   eidx     = (const int*)  d_in[4];
    const float* W_in_i   = (const float*)d_in[5];
    const float* b_in_i   = (const float*)d_in[6];
    const float* W_in_m   = (const float*)d_in[7];
    const float* b_in_m   = (const float*)d_in[8];
    const float* W_score  = (const float*)d_in[9];
    const float* b_score  = (const float*)d_in[10];
    const float* W_out_i  = (const float*)d_in[11];
    const float* b_out_i  = (const float*)d_in[12];
    const float* W_out_m  = (const float*)d_in[13];
    const float* b_out_m  = (const float*)d_in[14];
    float* out = (float*)d_out;

    // workspace layout (floats)
    float* ws = (float*)d_ws;
    size_t off = 0;
    float* xp_i  = ws + off; off += (size_t)N_INTT * FEAT_DIM;
    float* xp_m  = ws + off; off += (size_t)N_MVTX * FEAT_DIM;
    float* sum_i = ws + off; off += (size_t)N_INTT * EDGE_DIM;   // zero region start
    float* max_i = ws + off; off += (size_t)N_INTT * EDGE_DIM;
    float* cnt_i = ws + off; off += (size_t)N_INTT;
    float* sum_m = ws + off; off += (size_t)N_MVTX * EDGE_DIM;
    float* max_m = ws + off; off += (size_t)N_MVTX * EDGE_DIM;
    float* cnt_m = ws + off; off += (size_t)N_MVTX;

    const size_t zcount = off - 2 * (size_t)N_INTT * FEAT_DIM;   // sums+maxes+cnts
    zero_kernel<<<4096, 256, 0, stream>>>(sum_i, zcount);

    in_gemm<<<N_INTT / 16, 256, 0, stream>>>(x_intt, W_in_i, b_in_i, xp_i);
    in_gemm<<<N_MVTX / 16, 256, 0, stream>>>(x_mvtx, W_in_m, b_in_m, xp_m);

    edge_kernel<<<N_EDGES / 8, 256, 0, stream>>>(xp_i, xp_m, eidx, W_score, b_score,
                                                 sum_i, max_i, cnt_i,
                                                 sum_m, max_m, cnt_m);

    out_gemm<<<N_INTT / 16, 128, 0, stream>>>(x_intt, xp_i, sum_i, max_i, cnt_i,
                                              W_out_i, b_out_i, out);
    out_gemm<<<N_MVTX / 16, 128, 0, stream>>>(x_mvtx, xp_m, sum_m, max_m, cnt_m,
                                              W_out_m, b_out_m,
                                              out + (size_t)N_INTT * OUT_DIM);
}